// PGBF_58548994179774
// MI455X (gfx1250) — compile-verified
//
#include <hip/hip_runtime.h>
#include <hip/hip_bf16.h>
#include <math.h>

typedef __attribute__((ext_vector_type(2))) float v2f;
typedef __attribute__((ext_vector_type(8))) float v8f;

#define NN   8192
#define DIN  384
#define DD   512
#define TOPK 6

__device__ __forceinline__ v8f wmma4(v2f a, v2f b, v8f c) {
    // D = A(16x4,f32) * B(4x16,f32) + C(16x16,f32)
    return __builtin_amdgcn_wmma_f32_16x16x4_f32(
        /*neg_a=*/false, a, /*neg_b=*/false, b,
        /*c_mod=*/(short)0, c, /*reuse_a=*/false, /*reuse_b=*/false);
}

__device__ __forceinline__ float lrelu(float x) {
    return x > 0.0f ? x : 0.01f * x;
}

// ---------------------------------------------------------------------------
// Generic fused GEMM: out[M,N] = act(A[M,K] @ W[K,N] + bias[N])
// Block = 256 threads (8 waves), tile 64(M) x 128(N), K staged in 32-chunks.
// Waves: wm in 0..3 (16-row strips), wn in 0..1 (64-col strips, 4 subtiles).
// A and B both stored in LDS with the K index innermost and an EVEN padded
// stride (34): fragment .x/.y are adjacent + 8B aligned (one ds_load_b64,
// no v_mov fixups) and 16 consecutive lanes hit banks 2n mod 64 (conflict-
// free).  Requires M%64==0, N%128==0, K%32==0 (true for all uses here).
// ---------------------------------------------------------------------------
__global__ __launch_bounds__(256) void gemm_bias_act_kernel(
    const float* __restrict__ A, const float* __restrict__ W,
    const float* __restrict__ bias, float* __restrict__ out,
    int M, int K, int N, int act) {
    __shared__ float As[64 * 34];    // [m][k]
    __shared__ float Bs[128 * 34];   // [n][k]  (transposed stage)
    const int tid  = threadIdx.x;
    const int lane = tid & 31;
    const int wid  = tid >> 5;
    const int wm   = wid & 3;
    const int wn   = wid >> 2;
    const int li   = lane & 15;
    const int half = lane >> 4;
    const int mBase = blockIdx.y * 64;
    const int nBase = blockIdx.x * 128;

    v8f acc[4] = {{}, {}, {}, {}};

    for (int kb = 0; kb < K; kb += 32) {
        __syncthreads();
        #pragma unroll
        for (int idx = tid; idx < 64 * 32; idx += 256) {
            int m = idx >> 5, k = idx & 31;                 // coalesced over k
            As[m * 34 + k] = A[(size_t)(mBase + m) * K + kb + k];
        }
        #pragma unroll
        for (int idx = tid; idx < 128 * 32; idx += 256) {
            int n = idx & 127, k = idx >> 7;                // coalesced over n
            Bs[n * 34 + k] = W[(size_t)(kb + k) * N + nBase + n];
        }
        __syncthreads();
        #pragma unroll
        for (int kk = 0; kk < 32; kk += 4) {
            const int k0 = kk + 2 * half;
            v2f a;
            a.x = As[(wm * 16 + li) * 34 + k0];
            a.y = As[(wm * 16 + li) * 34 + k0 + 1];
            #pragma unroll
            for (int s = 0; s < 4; ++s) {
                const int c = wn * 64 + s * 16 + li;
                v2f b;
                b.x = Bs[c * 34 + k0];
                b.y = Bs[c * 34 + k0 + 1];
                acc[s] = wmma4(a, b, acc[s]);
            }
        }
    }

    #pragma unroll
    for (int s = 0; s < 4; ++s) {
        #pragma unroll
        for (int r = 0; r < 8; ++r) {
            const int row = mBase + wm * 16 + r + 8 * half;
            const int col = nBase + wn * 64 + s * 16 + li;
            float v = acc[s][r] + bias[col];
            if (act) v = lrelu(v);
            out[(size_t)row * N + col] = v;
        }
    }
}

// ---------------------------------------------------------------------------
// Column mean over rows: mean[c] = (1/M) * sum_r x[r,c]
// ---------------------------------------------------------------------------
__global__ void colmean_kernel(const float* __restrict__ x, float* __restrict__ mean,
                               int M, int N) {
    int c = blockIdx.x * blockDim.x + threadIdx.x;
    if (c >= N) return;
    float s = 0.0f;
    for (int r = 0; r < M; ++r) s += x[(size_t)r * N + c];
    mean[c] = s * (1.0f / (float)M);
}

// x = (x + mean[col]) * 0.5, in place
__global__ void smooth_kernel(float* __restrict__ x, const float* __restrict__ mean,
                              int total, int N) {
    for (int i = blockIdx.x * blockDim.x + threadIdx.x; i < total;
         i += gridDim.x * blockDim.x) {
        x[i] = (x[i] + mean[i % N]) * 0.5f;
    }
}

__global__ void add_kernel(float* __restrict__ a, const float* __restrict__ b, int total) {
    for (int i = blockIdx.x * blockDim.x + threadIdx.x; i < total;
         i += gridDim.x * blockDim.x) {
        a[i] = a[i] + b[i];
    }
}

// ---------------------------------------------------------------------------
// Fused attention logits + top-k(6) + softmax over kept logits.
// One block (128 threads = 4 waves) per 16 rows. e_h rows (pre-scaled) live in
// LDS; sweep all 8192 columns in 128-wide tiles of e_t^T; each wave owns two
// 16x16 WMMA tiles (A fragment reused); logits spill to LDS; 16 owner threads
// keep a sorted top-6 in registers (strict > preserves top_k tie order).
// ---------------------------------------------------------------------------
__global__ __launch_bounds__(128) void attn_topk_kernel(
    const float* __restrict__ eh, const float* __restrict__ et,
    float* __restrict__ tp, int* __restrict__ tix_out) {
    __shared__ float Ah[16 * 516];   // [r][k], stride 516: banks 4r mod 64 distinct
    __shared__ float Bs[128 * 66];   // [c][k], stride 66: fragment-contiguous
    __shared__ float Lg[16 * 132];
    const int tid  = threadIdx.x;
    const int lane = tid & 31;
    const int w    = tid >> 5;       // wave 0..3 -> 32-column strip
    const int li   = lane & 15;
    const int half = lane >> 4;
    const int rowBase = blockIdx.x * 16;
    const float scale = 0.04419417382415922f;   // 512^-0.5

    for (int idx = tid; idx < 16 * DD; idx += 128) {
        int r = idx >> 9, k = idx & (DD - 1);
        Ah[r * 516 + k] = eh[(size_t)(rowBase + r) * DD + k] * scale;
    }

    float tv[TOPK];
    int   ti[TOPK];
    #pragma unroll
    for (int j = 0; j < TOPK; ++j) { tv[j] = -3.4e38f; ti[j] = 0; }
    __syncthreads();

    for (int cb = 0; cb < NN; cb += 128) {
        v8f acc0 = {};
        v8f acc1 = {};
        for (int kb = 0; kb < DD; kb += 64) {
            __syncthreads();
            #pragma unroll
            for (int idx = tid; idx < 128 * 64; idx += 128) {
                int c = idx >> 6, k = idx & 63;                    // coalesced over k
                const size_t gaddr = (size_t)(cb + c) * DD + kb + k;
                Bs[c * 66 + k] = et[gaddr];
                if ((k & 31) == 0 && kb + 64 < DD)                 // warm next K chunk
                    __builtin_prefetch(&et[gaddr + 64], 0, 0);
            }
            __syncthreads();
            #pragma unroll
            for (int kk = 0; kk < 64; kk += 4) {
                const int ka  = kb + kk + 2 * half;
                const int kbs = kk + 2 * half;
                const int c0  = w * 32 + li;
                v2f a;
                a.x = Ah[li * 516 + ka];
                a.y = Ah[li * 516 + ka + 1];
                v2f b0, b1;
                b0.x = Bs[c0 * 66 + kbs];
                b0.y = Bs[c0 * 66 + kbs + 1];
                b1.x = Bs[(c0 + 16) * 66 + kbs];
                b1.y = Bs[(c0 + 16) * 66 + kbs + 1];
                acc0 = wmma4(a, b0, acc0);
                acc1 = wmma4(a, b1, acc1);
            }
        }
        #pragma unroll
        for (int r = 0; r < 8; ++r) {
            Lg[(r + 8 * half) * 132 + w * 32 + li]      = acc0[r];
            Lg[(r + 8 * half) * 132 + w * 32 + 16 + li] = acc1[r];
        }
        __syncthreads();
        if (tid < 16) {
            #pragma unroll 4
            for (int j = 0; j < 128; ++j) {
                float v = Lg[tid * 132 + j];
                if (v > tv[TOPK - 1]) {          // strict > keeps earliest index on ties
                    tv[TOPK - 1] = v;
                    ti[TOPK - 1] = cb + j;
                    #pragma unroll
                    for (int p = TOPK - 1; p > 0; --p) {
                        if (tv[p] > tv[p - 1]) {
                            float t = tv[p]; tv[p] = tv[p - 1]; tv[p - 1] = t;
                            int   q = ti[p]; ti[p] = ti[p - 1]; ti[p - 1] = q;
                        }
                    }
                }
            }
        }
        // next iteration's kb-loop barriers protect Lg/Bs reuse
    }

    if (tid < 16) {
        const int n = rowBase + tid;
        float m = tv[0];
        float e[TOPK], s = 0.0f;
        #pragma unroll
        for (int j = 0; j < TOPK; ++j) { e[j] = __expf(tv[j] - m); s += e[j]; }
        const float inv = 1.0f / s;
        #pragma unroll
        for (int j = 0; j < TOPK; ++j) {
            tp[n * TOPK + j]       = e[j] * inv;
            tix_out[n * TOPK + j]  = ti[j];
        }
    }
}

// ---------------------------------------------------------------------------
// Per-row neighbor aggregation: gather 6 e_t rows, tanh gate, softmax over k,
// emit s_in = e_h + e_Nh and b_in = e_h * e_Nh.  One wave per row.
// ---------------------------------------------------------------------------
__global__ __launch_bounds__(256) void neigh_kernel(
    const float* __restrict__ eh, const float* __restrict__ et,
    const float* __restrict__ tp, const int* __restrict__ tix,
    float* __restrict__ s_in, float* __restrict__ b_in) {
    const int lane = threadIdx.x & 31;
    const int wv   = threadIdx.x >> 5;
    const int n    = blockIdx.x * 8 + wv;

    float p[TOPK]; int id[TOPK];
    #pragma unroll
    for (int j = 0; j < TOPK; ++j) {
        p[j]  = tp[n * TOPK + j];
        id[j] = tix[n * TOPK + j];
    }
    float ehv[16], nb[TOPK][16], kw[TOPK];
    #pragma unroll
    for (int j = 0; j < TOPK; ++j) kw[j] = 0.0f;
    #pragma unroll
    for (int k = 0; k < 16; ++k) ehv[k] = eh[(size_t)n * DD + lane + 32 * k];

    #pragma unroll
    for (int j = 0; j < TOPK; ++j) {
        const float pj = p[j];
        const size_t base = (size_t)id[j] * DD;
        #pragma unroll
        for (int k = 0; k < 16; ++k) {
            float bv = et[base + lane + 32 * k];
            nb[j][k] = bv;
            float ehr  = pj * bv + (1.0f - pj) * ehv[k];
            float gate = tanhf(ehv[k] + ehr);
            kw[j] += bv * gate;
        }
    }
    #pragma unroll
    for (int j = 0; j < TOPK; ++j)
        #pragma unroll
        for (int off = 16; off > 0; off >>= 1)
            kw[j] += __shfl_xor(kw[j], off, 32);

    float m = kw[0];
    #pragma unroll
    for (int j = 1; j < TOPK; ++j) m = fmaxf(m, kw[j]);
    float q[TOPK], s = 0.0f;
    #pragma unroll
    for (int j = 0; j < TOPK; ++j) { q[j] = __expf(kw[j] - m); s += q[j]; }
    const float inv = 1.0f / s;

    #pragma unroll
    for (int k = 0; k < 16; ++k) {
        float eN = 0.0f;
        #pragma unroll
        for (int j = 0; j < TOPK; ++j) eN += q[j] * inv * nb[j][k];
        const int d = lane + 32 * k;
        s_in[(size_t)n * DD + d] = ehv[k] + eN;
        b_in[(size_t)n * DD + d] = ehv[k] * eN;
    }
}

// g[n] = dot(a_hid[n,:256], Wa2) + ba2 ; one wave per row
__global__ __launch_bounds__(256) void matvec_kernel(
    const float* __restrict__ ah, const float* __restrict__ w2,
    const float* __restrict__ b2, float* __restrict__ g) {
    const int lane = threadIdx.x & 31;
    const int wv   = threadIdx.x >> 5;
    const int n    = blockIdx.x * 8 + wv;
    float s = 0.0f;
    #pragma unroll
    for (int k = 0; k < 8; ++k) {
        const int d = lane + 32 * k;
        s += ah[(size_t)n * 256 + d] * w2[d];
    }
    #pragma unroll
    for (int off = 16; off > 0; off >>= 1) s += __shfl_xor(s, off, 32);
    if (lane == 0) g[n] = s + b2[0];
}

// red[0] = max(g), red[1] = sum exp(g - max)
__global__ __launch_bounds__(256) void softmax_stats_kernel(
    const float* __restrict__ g, int n, float* __restrict__ red) {
    __shared__ float sm[256];
    const int tid = threadIdx.x;
    float m = -3.4e38f;
    for (int i = tid; i < n; i += 256) m = fmaxf(m, g[i]);
    sm[tid] = m; __syncthreads();
    for (int s = 128; s > 0; s >>= 1) {
        if (tid < s) sm[tid] = fmaxf(sm[tid], sm[tid + s]);
        __syncthreads();
    }
    const float mx = sm[0];
    __syncthreads();
    float acc = 0.0f;
    for (int i = tid; i < n; i += 256) acc += __expf(g[i] - mx);
    sm[tid] = acc; __syncthreads();
    for (int s = 128; s > 0; s >>= 1) {
        if (tid < s) sm[tid] += sm[tid + s];
        __syncthreads();
    }
    if (tid == 0) { red[0] = mx; red[1] = sm[0]; }
}

// out[c] = (1/Z) * sum_n exp(g[n]-max) * h[n,c]
__global__ void readout_kernel(const float* __restrict__ h, const float* __restrict__ g,
                               const float* __restrict__ red, float* __restrict__ out) {
    const int c = blockIdx.x * blockDim.x + threadIdx.x;  // 0..511
    const float mx   = red[0];
    const float invZ = 1.0f / red[1];
    float acc = 0.0f;
    for (int r = 0; r < NN; ++r) acc += __expf(g[r] - mx) * h[(size_t)r * DD + c];
    out[c] = acc * invZ;
}

// ---------------------------------------------------------------------------
extern "C" void kernel_launch(void* const* d_in, const int* in_sizes, int n_in,
                              void* d_out, int out_size, void* d_ws, size_t ws_size,
                              hipStream_t stream) {
    const float* x_path = (const float*)d_in[0];
    const float* W1  = (const float*)d_in[1];  const float* b1  = (const float*)d_in[2];
    const float* Wh  = (const float*)d_in[3];  const float* bh  = (const float*)d_in[4];
    const float* Wt  = (const float*)d_in[5];  const float* bt  = (const float*)d_in[6];
    const float* Wl1 = (const float*)d_in[7];  const float* bl1 = (const float*)d_in[8];
    const float* Wl2 = (const float*)d_in[9];  const float* bl2 = (const float*)d_in[10];
    const float* Wa1 = (const float*)d_in[11]; const float* ba1 = (const float*)d_in[12];
    const float* Wa2 = (const float*)d_in[13]; const float* ba2 = (const float*)d_in[14];
    float* out = (float*)d_out;

    const size_t E = (size_t)NN * DD;          // 4,194,304 floats
    float* ws = (float*)d_ws;
    float* x    = ws;                          // [N,D]  (fc1 out, smoothed in place)
    float* ehb  = ws + E;                      // [N,D]  e_h   -> later t1 / h
    float* etb  = ws + 2 * E;                  // [N,D]  e_t   -> later t2
    float* s_in = ws + 3 * E;                  // [N,D]  sum-path input -> later a_hid
    float* b_in = ws + 4 * E;                  // [N,D]  bi-path input
    float* mean = ws + 5 * E;                  // [D]
    float* tp   = mean + DD;                   // [N,6]  top-k softmax probs
    int*   tix  = (int*)(tp + (size_t)NN * TOPK); // [N,6] indices
    float* g    = (float*)(tix + (size_t)NN * TOPK); // [N]
    float* red  = g + NN;                      // [2]

    // 1. x = lrelu(x_path @ W1 + b1)
    gemm_bias_act_kernel<<<dim3(DD / 128, NN / 64), 256, 0, stream>>>(
        x_path, W1, b1, x, NN, DIN, DD, 1);
    // 2-3. mean smoothing
    colmean_kernel<<<dim3(2), 256, 0, stream>>>(x, mean, NN, DD);
    smooth_kernel<<<dim3(512), 256, 0, stream>>>(x, mean, NN * DD, DD);
    // 4-5. e_h, e_t
    gemm_bias_act_kernel<<<dim3(DD / 128, NN / 64), 256, 0, stream>>>(
        x, Wh, bh, ehb, NN, DD, DD, 0);
    gemm_bias_act_kernel<<<dim3(DD / 128, NN / 64), 256, 0, stream>>>(
        x, Wt, bt, etb, NN, DD, DD, 0);
    // 6. fused logits + top-6 + softmax
    attn_topk_kernel<<<dim3(NN / 16), 128, 0, stream>>>(ehb, etb, tp, tix);
    // 7. neighbor aggregation
    neigh_kernel<<<dim3(NN / 8), 256, 0, stream>>>(ehb, etb, tp, tix, s_in, b_in);
    // 8-9. sum_emb -> t1 (reuse ehb), bi_emb -> t2 (reuse etb)
    gemm_bias_act_kernel<<<dim3(DD / 128, NN / 64), 256, 0, stream>>>(
        s_in, Wl1, bl1, ehb, NN, DD, DD, 1);
    gemm_bias_act_kernel<<<dim3(DD / 128, NN / 64), 256, 0, stream>>>(
        b_in, Wl2, bl2, etb, NN, DD, DD, 1);
    // 10. h = t1 + t2 (into ehb)
    add_kernel<<<dim3(512), 256, 0, stream>>>(ehb, etb, NN * DD);
    // 11. a_hid = lrelu(h @ Wa1 + ba1)   [N,256]  (reuse s_in)
    gemm_bias_act_kernel<<<dim3(256 / 128, NN / 64), 256, 0, stream>>>(
        ehb, Wa1, ba1, s_in, NN, DD, 256, 1);
    // 12. g = a_hid @ Wa2 + ba2
    matvec_kernel<<<dim3(NN / 8), 256, 0, stream>>>(s_in, Wa2, ba2, g);
    // 13-14. node softmax + weighted readout
    softmax_stats_kernel<<<dim3(1), 256, 0, stream>>>(g, NN, red);
    readout_kernel<<<dim3(2), 256, 0, stream>>>(ehb, g, red, out);
}